// SpatialEncoder_45655502356617
// MI455X (gfx1250) — compile-verified
//
#include <hip/hip_runtime.h>
#include <hip/hip_bf16.h>
#include <math.h>

// Problem constants (match reference).
#define N_NODES 8192
#define N_EDGES 262144
#define CIN     256
#define COUT    256
#define BATCH   4

typedef float v2f __attribute__((ext_vector_type(2)));
typedef float v8f __attribute__((ext_vector_type(8)));

// ---------------- Kernel 1: zero per-src edge counts -------------------------
__global__ void se_zero_counts(int* __restrict__ cnt) {
    int i = blockIdx.x * blockDim.x + threadIdx.x;   // grid sized exactly to N_NODES
    cnt[i] = 0;
}

// ---------------- Kernel 2: histogram of edge sources ------------------------
__global__ void se_count_edges(const long long* __restrict__ src, int* __restrict__ cnt) {
    int e = blockIdx.x * blockDim.x + threadIdx.x;   // grid sized exactly to N_EDGES
    atomicAdd(&cnt[(int)src[e]], 1);
}

// ---------------- Kernel 3: single-block exclusive scan + deg_inv ------------
__global__ void se_scan(const int* __restrict__ cnt,
                        int* __restrict__ row_off,
                        int* __restrict__ cursor,
                        float* __restrict__ deg_inv) {
    __shared__ int partial[256];
    __shared__ int pre[257];
    const int t = threadIdx.x;
    const int chunk = N_NODES / 256;   // 32
    int s = 0;
    for (int k = 0; k < chunk; ++k) s += cnt[t * chunk + k];
    partial[t] = s;
    __syncthreads();
    if (t == 0) {
        int acc = 0;
        for (int k = 0; k < 256; ++k) { pre[k] = acc; acc += partial[k]; }
        pre[256] = acc;
    }
    __syncthreads();
    int off = pre[t];
    for (int k = 0; k < chunk; ++k) {
        int idx = t * chunk + k;
        int c = cnt[idx];
        row_off[idx] = off;
        cursor[idx]  = off;
        deg_inv[idx] = 1.0f / (float)(c + 1);        // +1 self loop; clip is a no-op
        off += c;
    }
    if (t == 255) row_off[N_NODES] = off;            // == N_EDGES
}

// ---------------- Kernel 4: fill CSR adjacency (dst lists grouped by src) ----
__global__ void se_fill_csr(const long long* __restrict__ src,
                            const long long* __restrict__ dst,
                            int* __restrict__ cursor,
                            int* __restrict__ edge_dst) {
    int e = blockIdx.x * blockDim.x + threadIdx.x;   // grid sized exactly to N_EDGES
    int s = (int)src[e];
    int pos = atomicAdd(&cursor[s], 1);
    edge_dst[pos] = (int)dst[e];
}

// ---------------- Kernel 5: transpose weights + fuse biases ------------------
// Wt[n][k] = W[k][n] so B fragments become contiguous 8-byte loads.
__global__ void se_prep_w(const float* __restrict__ Ws, const float* __restrict__ Wn,
                          const float* __restrict__ bs, const float* __restrict__ bn,
                          float* __restrict__ Wts, float* __restrict__ Wtn,
                          float* __restrict__ bsum) {
    int idx = blockIdx.x * blockDim.x + threadIdx.x;  // grid sized exactly to 65536
    int k = idx >> 8, n = idx & 255;
    Wts[n * 256 + k] = Ws[k * 256 + n];
    Wtn[n * 256 + k] = Wn[k * 256 + n];
    if (idx < COUT) bsum[idx] = bs[idx] + bn[idx];
}

// ---------------- Kernel 6: gather aggregation y = A_hat @ x -----------------
// One wave per node; all 4 batches handled by the same wave so the adjacency
// list is read once.  Channels: lane*4 covers [0,128), +128 covers [128,256).
__global__ void se_aggregate(const float* __restrict__ x,
                             const int* __restrict__ row_off,
                             const int* __restrict__ edge_dst,
                             const float* __restrict__ deg_inv,
                             float* __restrict__ y) {
    const int wave = (blockIdx.x * blockDim.x + threadIdx.x) >> 5;  // == node id
    const int lane = threadIdx.x & 31;
    const int i = wave;

    const int c0 = lane * 4;
    float4 acc[BATCH][2];
#pragma unroll
    for (int b = 0; b < BATCH; ++b) {
        const float* p = x + ((size_t)(b * N_NODES + i)) * CIN;
        acc[b][0] = *(const float4*)(p + c0);
        acc[b][1] = *(const float4*)(p + 128 + c0);
    }
    const int beg = row_off[i], end = row_off[i + 1];
    for (int t = beg; t < end; ++t) {
        const int j = edge_dst[t];                    // uniform across the wave
#pragma unroll
        for (int b = 0; b < BATCH; ++b) {
            const float* p = x + ((size_t)(b * N_NODES + j)) * CIN;
            float4 u = *(const float4*)(p + c0);
            float4 v = *(const float4*)(p + 128 + c0);
            acc[b][0].x += u.x; acc[b][0].y += u.y; acc[b][0].z += u.z; acc[b][0].w += u.w;
            acc[b][1].x += v.x; acc[b][1].y += v.y; acc[b][1].z += v.z; acc[b][1].w += v.w;
        }
    }
    const float s = deg_inv[i];
#pragma unroll
    for (int b = 0; b < BATCH; ++b) {
        float* q = y + ((size_t)(b * N_NODES + i)) * CIN;
        float4 u = acc[b][0], v = acc[b][1];
        u.x *= s; u.y *= s; u.z *= s; u.w *= s;
        v.x *= s; v.y *= s; v.z *= s; v.w *= s;
        *(float4*)(q + c0)       = u;
        *(float4*)(q + 128 + c0) = v;
    }
}

// ---------------- Kernel 7: fused dual GEMM + bias + exact GELU (WMMA f32) ---
// Each wave owns a 64x32 output tile: 4 stacked M-tiles x 2 N-tiles, so each
// A fragment (x and y rows) feeds 4 WMMAs.  K-loop over 256 in steps of 4
// using V_WMMA_F32_16X16X4_F32 (16 WMMA / 12 b64-loads per step).
__global__ void se_gemm_bias_gelu(const float* __restrict__ x,
                                  const float* __restrict__ y,
                                  const float* __restrict__ Wts,   // [n][k] transposed
                                  const float* __restrict__ Wtn,   // [n][k] transposed
                                  const float* __restrict__ bsum,  // bself+bnei
                                  float* __restrict__ out) {
    const int wave = (blockIdx.x << 3) + (threadIdx.x >> 5); // 4096 waves exactly
    const int lane = threadIdx.x & 31;
    const int np = wave & 7;           // n-pair: 8 pairs of 2 n-tiles
    const int mg = wave >> 3;          // row group: 512 groups of 64 rows
    const int row0 = mg * 64;
    const int half = lane >> 4;        // K-half selector per ISA A/B layout
    const int mlan = lane & 15;        // M index within tile / N index for B
    const int n0 = np * 32 + mlan;     // first output channel owned by this lane
    const int n1 = n0 + 16;            // second output channel

    v8f acc0[4] = {};                  // 16x16 accumulators for n-tile 0
    v8f acc1[4] = {};                  // 16x16 accumulators for n-tile 1

    for (int k0 = 0; k0 < CIN; k0 += 4) {
        const int ka = k0 + half * 2;
        // B fragments: contiguous v2f from transposed weights.
        // half 0 lanes hold K=k0,k0+1 ; half 1 lanes hold K=k0+2,k0+3.
        v2f bs0 = *(const v2f*)(Wts + (size_t)n0 * CIN + ka);
        v2f bs1 = *(const v2f*)(Wts + (size_t)n1 * CIN + ka);
        v2f bn0 = *(const v2f*)(Wtn + (size_t)n0 * CIN + ka);
        v2f bn1 = *(const v2f*)(Wtn + (size_t)n1 * CIN + ka);
#pragma unroll
        for (int mt = 0; mt < 4; ++mt) {
            const int row = row0 + mt * 16 + mlan;
            v2f ax = *(const v2f*)(x + (size_t)row * CIN + ka);
            v2f ay = *(const v2f*)(y + (size_t)row * CIN + ka);
            acc0[mt] = __builtin_amdgcn_wmma_f32_16x16x4_f32(
                false, ax, false, bs0, (short)0, acc0[mt], false, false);
            acc0[mt] = __builtin_amdgcn_wmma_f32_16x16x4_f32(
                false, ay, false, bn0, (short)0, acc0[mt], false, false);
            acc1[mt] = __builtin_amdgcn_wmma_f32_16x16x4_f32(
                false, ax, false, bs1, (short)0, acc1[mt], false, false);
            acc1[mt] = __builtin_amdgcn_wmma_f32_16x16x4_f32(
                false, ay, false, bn1, (short)0, acc1[mt], false, false);
        }
    }

    const float bias0 = bsum[n0];
    const float bias1 = bsum[n1];
#pragma unroll
    for (int mt = 0; mt < 4; ++mt) {
#pragma unroll
        for (int r = 0; r < 8; ++r) {
            const int row = row0 + mt * 16 + r + 8 * half;   // C/D layout
            float v0 = acc0[mt][r] + bias0;
            float v1 = acc1[mt][r] + bias1;
            v0 = 0.5f * v0 * (1.0f + erff(v0 * 0.70710678118654752f));  // exact GELU
            v1 = 0.5f * v1 * (1.0f + erff(v1 * 0.70710678118654752f));
            out[(size_t)row * COUT + n0] = v0;
            out[(size_t)row * COUT + n1] = v1;
        }
    }
}

// ---------------------------------------------------------------------------
extern "C" void kernel_launch(void* const* d_in, const int* in_sizes, int n_in,
                              void* d_out, int out_size, void* d_ws, size_t ws_size,
                              hipStream_t stream) {
    const float*     x     = (const float*)d_in[0];           // [B,N,C]
    const long long* eidx  = (const long long*)d_in[1];       // [2,E] int64
    const float*     Wself = (const float*)d_in[2];           // [C,OUT]
    const float*     bsel  = (const float*)d_in[3];           // [OUT]
    const float*     Wnei  = (const float*)d_in[4];           // [C,OUT]
    const float*     bnei  = (const float*)d_in[5];           // [OUT]
    float*           out   = (float*)d_out;                   // [B,N,OUT]

    const long long* src = eidx;
    const long long* dst = eidx + N_EDGES;

    // Workspace layout.
    char* ws = (char*)d_ws;
    size_t off = 0;
    float* y       = (float*)(ws + off); off += (size_t)BATCH * N_NODES * CIN * sizeof(float);
    float* Wts     = (float*)(ws + off); off += (size_t)CIN * COUT * sizeof(float);
    float* Wtn     = (float*)(ws + off); off += (size_t)CIN * COUT * sizeof(float);
    float* bsum    = (float*)(ws + off); off += (size_t)COUT * sizeof(float);
    float* deg_inv = (float*)(ws + off); off += (size_t)N_NODES * sizeof(float);
    int*   row_off = (int*)  (ws + off); off += (size_t)(N_NODES + 1) * sizeof(int);
    int*   cursor  = (int*)  (ws + off); off += (size_t)N_NODES * sizeof(int);
    int*   cnt     = (int*)  (ws + off); off += (size_t)N_NODES * sizeof(int);
    int*   edge_dst= (int*)  (ws + off); off += (size_t)N_EDGES * sizeof(int);
    (void)ws_size; (void)in_sizes; (void)n_in; (void)out_size;

    se_zero_counts<<<N_NODES / 256, 256, 0, stream>>>(cnt);
    se_count_edges<<<N_EDGES / 256, 256, 0, stream>>>(src, cnt);
    se_scan<<<1, 256, 0, stream>>>(cnt, row_off, cursor, deg_inv);
    se_fill_csr<<<N_EDGES / 256, 256, 0, stream>>>(src, dst, cursor, edge_dst);
    se_prep_w<<<(CIN * COUT) / 256, 256, 0, stream>>>(Wself, Wnei, bsel, bnei, Wts, Wtn, bsum);
    se_aggregate<<<N_NODES / 8, 256, 0, stream>>>(x, row_off, edge_dst, deg_inv, y);
    // 512 row-groups x 8 n-pairs = 4096 waves = 512 blocks of 8 waves.
    se_gemm_bias_gelu<<<512, 256, 0, stream>>>(x, y, Wts, Wtn, bsum, out);
}